// EfficientCfCCell_23931557773776
// MI455X (gfx1250) — compile-verified
//
#include <hip/hip_runtime.h>
#include <math.h>

#define Bsz   512
#define INsz  256
#define Hsz   1024
#define CONC  1280                  // IN + H
#define WMAT  (CONC * Hsz)          // floats per dense weight matrix (1,310,720)

typedef __attribute__((ext_vector_type(2))) float v2f;
typedef __attribute__((ext_vector_type(8))) float v8f;

// ---------------------------------------------------------------------------
// Zero the dense-weight staging region (must happen every call: scatter adds).
// ---------------------------------------------------------------------------
__global__ __launch_bounds__(256) void zero4(float4* __restrict__ p, int n4) {
    int i = blockIdx.x * blockDim.x + threadIdx.x;
    if (i < n4) p[i] = make_float4(0.f, 0.f, 0.f, 0.f);
}

// ---------------------------------------------------------------------------
// Build X^T [CONC x B] from input [B x IN] and hx [B x H] so that WMMA
// A-fragment loads (lanes = M rows) are contiguous in memory.
// ---------------------------------------------------------------------------
__global__ __launch_bounds__(256) void build_xt(const float* __restrict__ in,
                                                const float* __restrict__ hx,
                                                float* __restrict__ XT) {
    int t = blockIdx.x * blockDim.x + threadIdx.x;   // [0, CONC*B)
    int j = t >> 9;          // concat index 0..1279
    int b = t & (Bsz - 1);   // batch   index 0..511
    float v = (j < INsz) ? in[b * INsz + j] : hx[b * Hsz + (j - INsz)];
    XT[j * Bsz + b] = v;
}

// ---------------------------------------------------------------------------
// Densify: for each edge, add w*sign into 4 dense [CONC x H] row-major mats.
// atomicAdd handles duplicate (src,dst) pairs exactly like scatter-add.
// ---------------------------------------------------------------------------
__global__ __launch_bounds__(256) void scatter_w(const int*   __restrict__ src,
                                                 const int*   __restrict__ dst,
                                                 const float* __restrict__ signs,
                                                 const float* __restrict__ w1,
                                                 const float* __restrict__ w2,
                                                 const float* __restrict__ wa,
                                                 const float* __restrict__ wb,
                                                 float* __restrict__ W, int E) {
    int e = blockIdx.x * blockDim.x + threadIdx.x;
    if (e >= E) return;
    float  sg   = signs[e];
    size_t base = (size_t)src[e] * Hsz + (size_t)dst[e];
    atomicAdd(&W[base],            w1[e] * sg);
    atomicAdd(&W[WMAT + base],     w2[e] * sg);
    atomicAdd(&W[2ull*WMAT + base], wa[e] * sg);
    atomicAdd(&W[3ull*WMAT + base], wb[e] * sg);
}

// ---------------------------------------------------------------------------
// Fused 4-way GEMM + CfC epilogue.
// Block = 128 threads = 4 waves. Wave w owns N-tile (blockIdx.x*4+w)*16,
// M rows blockIdx.y*32 .. +31 (two 16-row WMMA subtiles), all 4 weight mats.
// f32 WMMA: V_WMMA_F32_16X16X4_F32, K stepped by 4, 1280/4 = 320 steps,
// 8 independent accumulator chains per step.
// ---------------------------------------------------------------------------
__global__ __launch_bounds__(128) void gemm_fused(const float* __restrict__ XT,
                                                  const float* __restrict__ W,
                                                  const float* __restrict__ ts,
                                                  const float* __restrict__ b_ff1,
                                                  const float* __restrict__ b_ff2,
                                                  const float* __restrict__ b_ta,
                                                  const float* __restrict__ b_tb,
                                                  float* __restrict__ out) {
    const int lane = threadIdx.x & 31;
    const int wave = threadIdx.x >> 5;

    const int n0   = (blockIdx.x * 4 + wave) * 16;   // N tile base
    const int m0   = blockIdx.y * 32;                // M tile base (2 subtiles)
    const int mrow = lane & 15;                      // A: lane -> M row
    const int ncol = lane & 15;                      // B/C/D: lane -> N col
    const int kp   = (lane >> 4) << 1;               // lane-half -> K pair {0,2}

    v8f acc[2][4];
    #pragma unroll
    for (int s = 0; s < 2; ++s)
        #pragma unroll
        for (int l = 0; l < 4; ++l)
            acc[s][l] = (v8f){0.f,0.f,0.f,0.f,0.f,0.f,0.f,0.f};

    const float* xa0 = XT + (m0      + mrow);        // col of X^T for subtile 0
    const float* xa1 = XT + (m0 + 16 + mrow);        // col of X^T for subtile 1

    for (int k0 = 0; k0 < CONC; k0 += 4) {
        const int kr = k0 + kp;                      // this lane's K row pair
        // A fragments (16x4 f32): VGPR0 = K=kp, VGPR1 = K=kp+1 ; coalesced.
        v2f a0, a1;
        a0.x = xa0[(size_t)kr * Bsz];
        a0.y = xa0[(size_t)(kr + 1) * Bsz];
        a1.x = xa1[(size_t)kr * Bsz];
        a1.y = xa1[(size_t)(kr + 1) * Bsz];

        #pragma unroll
        for (int l = 0; l < 4; ++l) {
            const float* Wp = W + (size_t)l * WMAT + (size_t)kr * Hsz + n0 + ncol;
            v2f b;
            b.x = Wp[0];        // B row K=kp
            b.y = Wp[Hsz];      // B row K=kp+1
            acc[0][l] = __builtin_amdgcn_wmma_f32_16x16x4_f32(
                false, a0, false, b, (short)0, acc[0][l], false, false);
            acc[1][l] = __builtin_amdgcn_wmma_f32_16x16x4_f32(
                false, a1, false, b, (short)0, acc[1][l], false, false);
        }
    }

    // -------------------- fused CfC epilogue --------------------
    const int   ng  = n0 + ncol;
    const float bf1 = b_ff1[ng];
    const float bf2 = b_ff2[ng];
    const float bta = b_ta[ng];
    const float btb = b_tb[ng];
    const int   mhi = (lane >> 4) << 3;              // C/D: lane-half -> +8 rows

    #pragma unroll
    for (int s = 0; s < 2; ++s) {
        #pragma unroll
        for (int r = 0; r < 8; ++r) {
            const int   m   = m0 + s * 16 + r + mhi;
            const float tsv = ts[m];
            const float f1  = tanhf(acc[s][0][r] + bf1);
            const float f2  = tanhf(acc[s][1][r] + bf2);
            const float ta  = acc[s][2][r] + bta;
            const float tb  = acc[s][3][r] + btb;
            const float sg  = 1.f / (1.f + expf(-(ta * tsv + tb)));
            const float h   = f1 * (1.f - sg) + sg * f2;
            const size_t o  = (size_t)m * Hsz + ng;
            out[o]                         = h;   // tuple element 0
            out[(size_t)Bsz * Hsz + o]     = h;   // tuple element 1 (same tensor)
        }
    }
}

// ---------------------------------------------------------------------------
extern "C" void kernel_launch(void* const* d_in, const int* in_sizes, int n_in,
                              void* d_out, int out_size, void* d_ws, size_t ws_size,
                              hipStream_t stream) {
    const float* input = (const float*)d_in[0];
    const float* hx    = (const float*)d_in[1];
    const float* ts    = (const float*)d_in[2];
    const int*   src   = (const int*)  d_in[3];
    const int*   dst   = (const int*)  d_in[4];
    const float* signs = (const float*)d_in[5];
    const float* w1    = (const float*)d_in[6];
    const float* w2    = (const float*)d_in[7];
    const float* wa    = (const float*)d_in[8];
    const float* wb    = (const float*)d_in[9];
    const float* bf1   = (const float*)d_in[10];
    const float* bf2   = (const float*)d_in[11];
    const float* bta   = (const float*)d_in[12];
    const float* btb   = (const float*)d_in[13];
    const int    E     = in_sizes[3];

    float* W  = (float*)d_ws;                 // 4 * WMAT floats (~21 MB)
    float* XT = W + 4ull * WMAT;              // CONC * B floats (~2.6 MB)
    float* out = (float*)d_out;

    // 1) zero dense weights (float4 stores)
    const int n4 = (4 * WMAT) / 4;
    zero4<<<(n4 + 255) / 256, 256, 0, stream>>>((float4*)W, n4);

    // 2) build X^T
    build_xt<<<(CONC * Bsz) / 256, 256, 0, stream>>>(input, hx, XT);

    // 3) densify sparse weights
    scatter_w<<<(E + 255) / 256, 256, 0, stream>>>(src, dst, signs,
                                                   w1, w2, wa, wb, W, E);

    // 4) fused 4-way WMMA GEMM + CfC epilogue
    dim3 grid(Hsz / 64, Bsz / 32);            // (16, 16) blocks x 4 waves
    gemm_fused<<<grid, 128, 0, stream>>>(XT, W, ts, bf1, bf2, bta, btb, out);
}